// HomoGraphEncoder_61881888801446
// MI455X (gfx1250) — compile-verified
//
#include <hip/hip_runtime.h>

// ---------------------------------------------------------------------------
// GAT (2 layers) + mean readout for MI455X (gfx1250, wave32).
// GEMMs: V_WMMA_F32_16X16X4_F32, register-blocked 16x64 per wave (A-frag
// reused across 4 WMMAs -> 4x less A traffic, better wmma:vmem ratio).
// Edge softmax/scatter: one wave per edge (grid-stride), lane-strided
// coalesced gathers + f32 atomic adds, with global_prefetch_b8 of the next
// edge's feature row to hide L2 gather latency behind the atomic pipeline.
// All node-feature arrays are << 192MB L2, so the edge phase is L2-resident.
// ---------------------------------------------------------------------------

typedef __attribute__((ext_vector_type(2))) float v2f;
typedef __attribute__((ext_vector_type(8))) float v8f;

#define NNODES 20000
#define NEDGES 320000
#define NGRAPH 16

// ---- workspace arena (bytes) ----
#define SZ_F1   (20000L * 384 * 4)        // 30,720,000  f1, later reused as h
#define OFF_F1  0L
#define OFF_NUM1 (OFF_F1 + SZ_F1)          // num1 (30.72MB); later f2 + num2
#define OFF_F2   OFF_NUM1                  // f2: 20000*128*4 = 10,240,000
#define OFF_NUM2 (OFF_NUM1 + 20000L*128*4) // num2: 10,240,000 (inside num1 region)
#define OFF_E    (OFF_NUM1 + SZ_F1)        // e attn: E*3*4 = 3,840,000 (e2 reuses)
#define OFF_M    (OFF_E + 320000L*3*4)     // m enc/val: N*3*4 (m2 reuses)
#define OFF_DEN  (OFF_M + 20000L*3*4)      // denom: N*3*4
#define OFF_EL   (OFF_DEN + 20000L*3*4)    // el: N*3*4
#define OFF_ER   (OFF_EL + 20000L*3*4)     // er: N*3*4
#define OFF_SUMS (OFF_ER + 20000L*3*4)     // sums: 16*128*4
#define OFF_CNTS (OFF_SUMS + 16L*128*4)    // cnts: 16*4 (contiguous after sums)
#define OFF_HG   (OFF_CNTS + 256L)         // hg: 16*128*4

// ---- ordered-uint encoding for float atomic max ----
__device__ __forceinline__ unsigned ford_enc(float f) {
  unsigned u = __float_as_uint(f);
  return (u & 0x80000000u) ? ~u : (u | 0x80000000u);
}
__device__ __forceinline__ float ford_dec(unsigned u) {
  return (u & 0x80000000u) ? __uint_as_float(u & 0x7FFFFFFFu)
                           : __uint_as_float(~u);
}
#define NEG_INF_ENC 0x007FFFFFu   // ford_enc(-inf)

// ---------------------------------------------------------------------------
__global__ void k_zero_f32(float* p, long n) {
  long i = (long)blockIdx.x * blockDim.x + threadIdx.x;
  if (i < n) p[i] = 0.0f;
}
__global__ void k_fill_u32(unsigned* p, unsigned v, long n) {
  long i = (long)blockIdx.x * blockDim.x + threadIdx.x;
  if (i < n) p[i] = v;
}

// ---------------------------------------------------------------------------
// C[M,N] = A[M,K] @ B[K,N], row-major f32. One wave per 16x64 C strip
// (4 accumulators); A fragment loaded once per k-step, used by 4 WMMAs.
// Layouts per ISA 7.12.2: A(16x4): lanes 0-15 -> K=(k,k+1) of row M=lane,
// lanes 16-31 -> K=(k+2,k+3) of row M=lane-16. B(4x16) mirrored.
// D: VGPR r -> row +r (lanes 0-15) / +r+8 (lanes 16-31), col = (lane&15).
__global__ void k_gemm_wmma_f32(const float* __restrict__ A,
                                const float* __restrict__ B,
                                float* __restrict__ C,
                                int M, int N, int K) {
  int wave = (int)((blockIdx.x * blockDim.x + threadIdx.x) >> 5);
  int lane = threadIdx.x & 31;
  int ntile = N >> 6;                    // 64-wide strips
  int tm = wave / ntile, tn = wave - tm * ntile;
  if (tm * 16 >= M) return;              // wave-uniform guard (EXEC stays all-1)
  int hi   = lane >> 4;
  int m    = tm * 16 + (lane & 15);
  int cb   = tn * 64 + (lane & 15);
  int kofs = hi << 1;
  const float* arow = A + (long)m * K;
  v8f a0 = {}, a1 = {}, a2 = {}, a3 = {};
  for (int k = 0; k < K; k += 4) {
    v2f a;
    a[0] = arow[k + kofs];
    a[1] = arow[k + kofs + 1];
    const float* br0 = B + (long)(k + kofs)     * N + cb;
    const float* br1 = B + (long)(k + kofs + 1) * N + cb;
    v2f b0, b1, b2, b3;
    b0[0] = br0[0];  b0[1] = br1[0];
    b1[0] = br0[16]; b1[1] = br1[16];
    b2[0] = br0[32]; b2[1] = br1[32];
    b3[0] = br0[48]; b3[1] = br1[48];
    a0 = __builtin_amdgcn_wmma_f32_16x16x4_f32(false, a, false, b0, (short)0, a0, false, false);
    a1 = __builtin_amdgcn_wmma_f32_16x16x4_f32(false, a, false, b1, (short)0, a1, false, false);
    a2 = __builtin_amdgcn_wmma_f32_16x16x4_f32(false, a, false, b2, (short)0, a2, false, false);
    a3 = __builtin_amdgcn_wmma_f32_16x16x4_f32(false, a, false, b3, (short)0, a3, false, false);
  }
  int rbase = tm * 16 + (hi ? 8 : 0);
  for (int r = 0; r < 8; ++r) {
    float* crow = C + (long)(rbase + r) * N + cb;
    crow[0]  = a0[r];
    crow[16] = a1[r];
    crow[32] = a2[r];
    crow[48] = a3[r];
  }
}

// ---------------------------------------------------------------------------
// el[n,h] = dot(f[n,h,:], al[h,:]);  er likewise.
__global__ void k_elr(const float* __restrict__ f, const float* __restrict__ al,
                      const float* __restrict__ ar, float* __restrict__ el,
                      float* __restrict__ er, int N, int H, int D) {
  int i = blockIdx.x * blockDim.x + threadIdx.x;   // i = n*H + h
  if (i >= N * H) return;
  int n = i / H, h = i - n * H;
  const float* fp  = f + (long)n * H * D + (long)h * D;
  const float* alp = al + (long)h * D;
  const float* arp = ar + (long)h * D;
  float sl = 0.f, sr = 0.f;
  for (int d = 0; d < D; ++d) { float v = fp[d]; sl += v * alp[d]; sr += v * arp[d]; }
  el[i] = sl; er[i] = sr;
}

// e = leaky_relu(el[src]+er[dst]); segment-max into menc (ordered-uint).
__global__ void k_edge_max(const float* __restrict__ el, const float* __restrict__ er,
                           const int* __restrict__ src, const int* __restrict__ dst,
                           float* __restrict__ eatt, unsigned* __restrict__ menc,
                           int E, int H) {
  int i = blockIdx.x * blockDim.x + threadIdx.x;   // i = e*H + h
  if (i >= E * H) return;
  int e = i / H, h = i - e * H;
  int s = src[e], d = dst[e];
  float v = el[s * H + h] + er[d * H + h];
  v = v > 0.0f ? v : 0.2f * v;
  eatt[i] = v;
  atomicMax(&menc[d * H + h], ford_enc(v));
}

// decode menc in place to float; non-finite (no incoming edges) -> 0.
__global__ void k_decode_m(unsigned* __restrict__ m, long n) {
  long i = (long)blockIdx.x * blockDim.x + threadIdx.x;
  if (i >= n) return;
  float v = ford_dec(m[i]);
  if (__builtin_isinf(v)) v = 0.0f;
  ((float*)m)[i] = v;
}

// Grid-stride, one wave per edge: ex=exp(e-m[dst]); denom += ex;
// num[dst] += ex * f[src].  D == 128 for both layers (col>>7 = head).
// While the current edge's atomics drain, prefetch the next edge's source
// feature row (global_prefetch_b8): lanes spaced HD/32 floats apart touch
// every 128B line of the HD*4-byte row.
__global__ void k_edge_accum(const float* __restrict__ eatt,
                             const float* __restrict__ mval,
                             const float* __restrict__ f,
                             const int* __restrict__ src,
                             const int* __restrict__ dst,
                             float* __restrict__ denom, float* __restrict__ num,
                             int E, int H) {
  int gwave  = (int)((blockIdx.x * blockDim.x + threadIdx.x) >> 5);
  int lane   = threadIdx.x & 31;
  int stride = (int)((gridDim.x * blockDim.x) >> 5);
  int HD = H << 7;                       // H * 128
  for (int e = gwave; e < E; e += stride) {
    int s = src[e], d = dst[e];
    int en = e + stride;                 // wave-uniform
    if (en < E) {
      int sn = src[en];
      __builtin_prefetch(f + (long)sn * HD + lane * (HD >> 5), 0, 3);
    }
    float ex[3];
    for (int h = 0; h < H; ++h)
      ex[h] = __expf(eatt[e * H + h] - mval[d * H + h]);
    if (lane < H) atomicAdd(&denom[d * H + lane], ex[lane]);
    const float* fs = f + (long)s * HD;
    float* nd = num + (long)d * HD;
    for (int c = lane; c < HD; c += 32)
      atomicAdd(&nd[c], ex[c >> 7] * fs[c]);
  }
}

// out = elu(num/denom + bias)  (layer-1 epilogue; writes h over f1 buffer)
__global__ void k_finalize_elu(const float* __restrict__ num,
                               const float* __restrict__ denom,
                               const float* __restrict__ bias,
                               float* __restrict__ out, int N, int H) {
  long i = (long)blockIdx.x * blockDim.x + threadIdx.x;
  int HD = H << 7;
  if (i >= (long)N * HD) return;
  long n = i / HD; int c = (int)(i - n * HD); int h = c >> 7;
  float dn = denom[n * H + h];
  float v = num[i] / (dn > 0.0f ? dn : 1.0f) + bias[c];
  out[i] = v > 0.0f ? v : (__expf(v) - 1.0f);
}

// layer-2 epilogue + graph readout: sums[gid[n]] += num/denom + bias
__global__ void k_finalize_readout(const float* __restrict__ num,
                                   const float* __restrict__ denom,
                                   const float* __restrict__ bias,
                                   const int* __restrict__ gid,
                                   float* __restrict__ sums, int N, int D) {
  long i = (long)blockIdx.x * blockDim.x + threadIdx.x;
  if (i >= (long)N * D) return;
  int n = (int)(i / D), c = (int)(i - (long)n * D);
  float dn = denom[n];
  float v = num[i] / (dn > 0.0f ? dn : 1.0f) + bias[c];
  atomicAdd(&sums[gid[n] * D + c], v);
}

__global__ void k_count(const int* __restrict__ gid, float* __restrict__ cnts, int N) {
  int i = blockIdx.x * blockDim.x + threadIdx.x;
  if (i < N) atomicAdd(&cnts[gid[i]], 1.0f);
}

__global__ void k_mean(const float* __restrict__ sums, const float* __restrict__ cnts,
                       float* __restrict__ hg, int G, int D) {
  int i = blockIdx.x * blockDim.x + threadIdx.x;
  if (i >= G * D) return;
  float c = cnts[i / D];
  hg[i] = sums[i] / (c > 1.0f ? c : 1.0f);
}

// out[16,128] = relu(hg[16,128] @ linW[128,128] + linb). One block, 8 waves,
// each wave owns one 16x16 N-tile; K=128 via 32 f32 WMMA steps.
__global__ void k_final_gemm_relu(const float* __restrict__ A,
                                  const float* __restrict__ B,
                                  const float* __restrict__ bias,
                                  float* __restrict__ out) {
  const int Kd = 128, Nc = 128;
  int wave = threadIdx.x >> 5;           // 0..7 = tn
  int lane = threadIdx.x & 31;
  int hi   = lane >> 4;
  int m    = lane & 15;
  int cb   = wave * 16 + (lane & 15);
  int kofs = hi << 1;
  v8f acc = {};
  for (int k = 0; k < Kd; k += 4) {
    v2f a, b;
    a[0] = A[m * Kd + k + kofs];
    a[1] = A[m * Kd + k + kofs + 1];
    b[0] = B[(k + kofs)     * Nc + cb];
    b[1] = B[(k + kofs + 1) * Nc + cb];
    acc = __builtin_amdgcn_wmma_f32_16x16x4_f32(false, a, false, b,
                                                (short)0, acc, false, false);
  }
  int rbase = hi ? 8 : 0;
  float bb = bias[cb];
  for (int r = 0; r < 8; ++r) {
    float v = acc[r] + bb;
    out[(rbase + r) * Nc + cb] = v > 0.0f ? v : 0.0f;
  }
}

// ---------------------------------------------------------------------------
static inline int nblk(long n, int bs) { return (int)((n + bs - 1) / bs); }

extern "C" void kernel_launch(void* const* d_in, const int* in_sizes, int n_in,
                              void* d_out, int out_size, void* d_ws, size_t ws_size,
                              hipStream_t stream) {
  (void)in_sizes; (void)n_in; (void)out_size; (void)ws_size;
  const float* x     = (const float*)d_in[0];   // [20000,128]
  const int*   src   = (const int*)  d_in[1];   // [320000]
  const int*   dst   = (const int*)  d_in[2];   // [320000]
  const int*   gid   = (const int*)  d_in[3];   // [20000]
  const float* W1    = (const float*)d_in[4];   // [128,384]
  const float* al1   = (const float*)d_in[5];   // [3,128]
  const float* ar1   = (const float*)d_in[6];   // [3,128]
  const float* b1    = (const float*)d_in[7];   // [384]
  const float* W2    = (const float*)d_in[8];   // [384,128]
  const float* al2   = (const float*)d_in[9];   // [1,128]
  const float* ar2   = (const float*)d_in[10];  // [1,128]
  const float* b2    = (const float*)d_in[11];  // [128]
  const float* linW  = (const float*)d_in[12];  // [128,128]
  const float* linb  = (const float*)d_in[13];  // [128]
  float* out = (float*)d_out;                   // [16,128]

  char* ws = (char*)d_ws;
  float*    f1    = (float*)   (ws + OFF_F1);    // later h
  float*    num1  = (float*)   (ws + OFF_NUM1);
  float*    f2    = (float*)   (ws + OFF_F2);
  float*    num2  = (float*)   (ws + OFF_NUM2);
  float*    eatt  = (float*)   (ws + OFF_E);
  unsigned* menc  = (unsigned*)(ws + OFF_M);
  float*    mval  = (float*)   (ws + OFF_M);
  float*    den   = (float*)   (ws + OFF_DEN);
  float*    el    = (float*)   (ws + OFF_EL);
  float*    er    = (float*)   (ws + OFF_ER);
  float*    sums  = (float*)   (ws + OFF_SUMS);
  float*    cnts  = (float*)   (ws + OFF_CNTS);
  float*    hg    = (float*)   (ws + OFF_HG);

  const int BS = 256;
  const int ACC_BLOCKS = 4096;           // 32768 waves, ~10 edges each

  // ---- layer 1 init ----
  k_zero_f32 <<<nblk(20000L*384, BS), BS, 0, stream>>>(num1, 20000L*384);
  k_fill_u32 <<<nblk(20000L*3,   BS), BS, 0, stream>>>(menc, NEG_INF_ENC, 20000L*3);
  k_zero_f32 <<<nblk(20000L*3,   BS), BS, 0, stream>>>(den, 20000L*3);
  k_zero_f32 <<<nblk(2048L+64,   BS), BS, 0, stream>>>(sums, 2048L+64); // sums+cnts

  // ---- layer 1 ----
  // f1 = x @ W1 : 1250 x 6 strips (16x64) = 7500 waves
  k_gemm_wmma_f32<<<nblk(7500L*32, BS), BS, 0, stream>>>(x, W1, f1, NNODES, 384, 128);
  k_elr          <<<nblk(20000L*3, BS), BS, 0, stream>>>(f1, al1, ar1, el, er, NNODES, 3, 128);
  k_edge_max     <<<nblk(320000L*3, BS), BS, 0, stream>>>(el, er, src, dst, eatt, menc, NEDGES, 3);
  k_decode_m     <<<nblk(20000L*3, BS), BS, 0, stream>>>(menc, 20000L*3);
  k_edge_accum   <<<ACC_BLOCKS, BS, 0, stream>>>(eatt, mval, f1, src, dst, den, num1, NEDGES, 3);
  k_finalize_elu <<<nblk(20000L*384, BS), BS, 0, stream>>>(num1, den, b1, f1, NNODES, 3); // h -> f1 buf

  // ---- layer 2 ----
  // f2 = h @ W2 : 1250 x 2 strips = 2500 waves (writes over dead num1)
  k_gemm_wmma_f32<<<nblk(2500L*32, BS), BS, 0, stream>>>(f1, W2, f2, NNODES, 128, 384);
  k_elr          <<<nblk(20000L, BS), BS, 0, stream>>>(f2, al2, ar2, el, er, NNODES, 1, 128);
  k_fill_u32     <<<nblk(20000L, BS), BS, 0, stream>>>(menc, NEG_INF_ENC, 20000L);
  k_zero_f32     <<<nblk(20000L, BS), BS, 0, stream>>>(den, 20000L);
  k_zero_f32     <<<nblk(20000L*128, BS), BS, 0, stream>>>(num2, 20000L*128);
  k_edge_max     <<<nblk(320000L, BS), BS, 0, stream>>>(el, er, src, dst, eatt, menc, NEDGES, 1);
  k_decode_m     <<<nblk(20000L, BS), BS, 0, stream>>>(menc, 20000L);
  k_edge_accum   <<<ACC_BLOCKS, BS, 0, stream>>>(eatt, mval, f2, src, dst, den, num2, NEDGES, 1);

  // ---- readout ----
  k_count           <<<nblk(20000L, BS), BS, 0, stream>>>(gid, cnts, NNODES);
  k_finalize_readout<<<nblk(20000L*128, BS), BS, 0, stream>>>(num2, den, b2, gid, sums, NNODES, 128);
  k_mean            <<<nblk(2048L, BS), BS, 0, stream>>>(sums, cnts, hg, NGRAPH, 128);
  k_final_gemm_relu <<<1, BS, 0, stream>>>(hg, linW, linb, out);
}